// RPool_76141180223557
// MI455X (gfx1250) — compile-verified
//
#include <hip/hip_runtime.h>
#include <hip/hip_bf16.h>

typedef __attribute__((ext_vector_type(2))) float v2f;
typedef __attribute__((ext_vector_type(4))) float v4f;
typedef __attribute__((ext_vector_type(8))) float v8f;

#define THREADS 256
#define CHUNK   4096
#define PER_T   16

// order-preserving map: larger float -> larger unsigned key
__device__ __forceinline__ unsigned fkey(float s) {
  unsigned u = __float_as_uint(s);
  return (u & 0x80000000u) ? ~u : (u | 0x80000000u);
}

// ---------------------------------------------------------------------------
// scores = x @ W + b via V_WMMA_F32_16X16X4_F32. One wave computes 16 rows:
// A tile = 16x4 f32 slice of x, B tile = W[k..k+3] broadcast to all 16 cols,
// chained accumulation over K=128 in steps of 4 (32 WMMAs). f32 in/out keeps
// exact precision for the subsequent exact top-k.
// A layout: lane l (m=l&15, hi=l>>4): vgpr0=A[m,2hi], vgpr1=A[m,2hi+1]
// B layout: lane l: vgpr0=B[2hi,n], vgpr1=B[2hi+1,n]  (constant over n here)
// D layout: lane 0 holds D[0..7,0] in c[0..7]; lane 16 holds D[8..15,0].
// ---------------------------------------------------------------------------
__global__ __launch_bounds__(256) void k_scores(const float* __restrict__ x,
                                                const float* __restrict__ W,
                                                const float* __restrict__ bias,
                                                float* __restrict__ scores,
                                                int N) {
  int lane = threadIdx.x & 31;
  int wave = (int)((blockIdx.x * (blockDim.x >> 5)) + (threadIdx.x >> 5));
  int rowBase = wave * 16;
  if (rowBase >= N) return;                 // wave-uniform exit
  int m  = lane & 15;
  int hi = lane >> 4;
  int row = rowBase + m;
  if (row >= N) row = N - 1;                // clamp tail (stores guarded below)
  const float* xr = x + (size_t)row * 128;
  v8f c = {0.f,0.f,0.f,0.f,0.f,0.f,0.f,0.f};
  #pragma unroll
  for (int k0 = 0; k0 < 128; k0 += 4) {
    int kk = k0 + 2 * hi;
    v2f a; a.x = xr[kk]; a.y = xr[kk + 1];
    v2f bm; bm.x = W[kk]; bm.y = W[kk + 1];
    c = __builtin_amdgcn_wmma_f32_16x16x4_f32(false, a, false, bm,
                                              (short)0, c, false, false);
  }
  float bb = bias[0];
  if (m == 0) {                             // lanes 0 and 16
    int base = rowBase + 8 * hi;
    #pragma unroll
    for (int r = 0; r < 8; ++r) {
      int rr = base + r;
      if (rr < N) scores[rr] = c[r] + bb;
    }
  }
}

// ---------------------------------------------------------------------------
// init: zero radix histogram + control block; ctl[1] = k (remaining count)
// ---------------------------------------------------------------------------
__global__ void k_init(unsigned* hist, unsigned* ctl, unsigned kk) {
  int t = blockIdx.x * blockDim.x + threadIdx.x;
  if (t < 2048) hist[t] = 0u;
  if (t < 16)   ctl[t] = (t == 1) ? kk : 0u;
}

// ---------------------------------------------------------------------------
// radix-select histogram pass (pass 0: key[31:21], 1: key[20:10], 2: key[9:0])
// LDS privatized histogram, flushed with global atomics.
// ---------------------------------------------------------------------------
__global__ __launch_bounds__(256) void k_hist(const float* __restrict__ scores,
                                              unsigned* __restrict__ hist,
                                              const unsigned* __restrict__ ctl,
                                              int N, int pass) {
  __shared__ unsigned h[2048];
  int t = threadIdx.x;
  for (int j = t; j < 2048; j += 256) h[j] = 0u;
  __syncthreads();
  unsigned pref = ctl[0];
  int stride = gridDim.x * blockDim.x;
  for (int i = blockIdx.x * blockDim.x + t; i < N; i += stride) {
    unsigned key = fkey(scores[i]);
    bool act; unsigned dig;
    if (pass == 0)      { act = true;                    dig = key >> 21; }
    else if (pass == 1) { act = ((key >> 21) == pref);   dig = (key >> 10) & 2047u; }
    else                { act = ((key >> 10) == pref);   dig = key & 1023u; }
    if (act) atomicAdd(&h[dig], 1u);
  }
  __syncthreads();
  for (int j = t; j < 2048; j += 256)
    if (h[j]) atomicAdd(&hist[j], h[j]);
}

// ---------------------------------------------------------------------------
// single-block descending radix scan: find digit d* = max d with
// suffix_count(d) >= k_remaining; update prefix / k_remaining; on the last
// pass emit the exact threshold key (ctl[5]) and tie budget (ctl[6]).
// Also zeroes the histogram for the next pass.
// ---------------------------------------------------------------------------
__global__ __launch_bounds__(1024) void k_scan_radix(unsigned* hist, unsigned* ctl,
                                                     int nbins, int bits, int last) {
  __shared__ unsigned arr[2048];   // reversed counts: arr[j] = hist[nbins-1-j]
  __shared__ unsigned ps[1024];    // inclusive scan of pair sums
  int t = threadIdx.x;
  int pairs = nbins >> 1;
  for (int j = t; j < nbins; j += 1024) arr[j] = hist[nbins - 1 - j];
  __syncthreads();
  for (int j = t; j < nbins; j += 1024) hist[j] = 0u;   // reset for next pass
  unsigned a0 = 0u, a1 = 0u, psum = 0u;
  if (t < pairs) { a0 = arr[2 * t]; a1 = arr[2 * t + 1]; psum = a0 + a1; }
  ps[t] = psum;
  __syncthreads();
  for (int o = 1; o < pairs; o <<= 1) {
    unsigned v = (t >= o && t < pairs) ? ps[t - o] : 0u;
    __syncthreads();
    if (t < pairs) ps[t] += v;
    __syncthreads();
  }
  unsigned krem = ctl[1];
  if (t < pairs) {
    unsigned pre1 = ps[t];                         // prefix incl. of rev idx 2t+1
    unsigned pre0 = ps[t] - a1;                    // prefix incl. of rev idx 2t
    unsigned psm1 = (t > 0) ? ps[t - 1] : 0u;      // prefix incl. of rev idx 2t-1
    int jstar = -1; unsigned higher = 0u;
    if (pre0 >= krem && (t == 0 || psm1 < krem)) { jstar = 2 * t;     higher = psm1; }
    else if (pre1 >= krem && pre0 < krem)        { jstar = 2 * t + 1; higher = pre0; }
    if (jstar >= 0) {                              // exactly one thread matches
      unsigned d = (unsigned)(nbins - 1 - jstar);
      unsigned newpref = (ctl[0] << bits) | d;
      ctl[0] = newpref;
      ctl[2] += higher;                            // cumulative strictly-above
      unsigned nk = krem - higher;
      ctl[1] = nk;
      if (last) { ctl[5] = newpref; ctl[6] = nk; } // threshold key, tie budget
    }
  }
}

// ---------------------------------------------------------------------------
// keep-mask: key > threshold always kept; ties at the exact threshold key
// consume the tie budget (exact-equal float ties are measure-zero).
// ---------------------------------------------------------------------------
__global__ void k_mask(const float* __restrict__ scores, unsigned* __restrict__ mask,
                       unsigned* ctl, int N) {
  int i = blockIdx.x * blockDim.x + threadIdx.x;
  if (i >= N) return;
  unsigned key = fkey(scores[i]);
  unsigned th = ctl[5], ties = ctl[6];
  unsigned keep = 0u;
  if (key > th) keep = 1u;
  else if (key == th) keep = (atomicAdd(&ctl[3], 1u) < ties) ? 1u : 0u;
  mask[i] = keep;
}

// --------------------- stable compaction helpers (count/scan/scatter) -------
__global__ __launch_bounds__(256) void k_ncount(const unsigned* __restrict__ mask,
                                                unsigned* bcnt, int N) {
  __shared__ unsigned red[THREADS];
  int t = threadIdx.x, base = blockIdx.x * CHUNK + t * PER_T;
  unsigned c = 0u;
  for (int j = 0; j < PER_T; ++j) { int i = base + j; if (i < N) c += mask[i]; }
  red[t] = c; __syncthreads();
  for (int s = THREADS / 2; s > 0; s >>= 1) { if (t < s) red[t] += red[t + s]; __syncthreads(); }
  if (t == 0) bcnt[blockIdx.x] = red[0];
}

__global__ __launch_bounds__(1024) void k_scan_blocks(const unsigned* __restrict__ cnt,
                                                      unsigned* off, int nb,
                                                      unsigned* total_ws, float* total_out) {
  __shared__ unsigned s[1024];
  int t = threadIdx.x;
  unsigned v = (t < nb) ? cnt[t] : 0u;
  s[t] = v; __syncthreads();
  for (int o = 1; o < 1024; o <<= 1) {
    unsigned u = (t >= o) ? s[t - o] : 0u;
    __syncthreads();
    s[t] += u;
    __syncthreads();
  }
  if (t < nb) off[t] = s[t] - v;                       // exclusive
  if (t == 1023) {
    if (total_ws)  total_ws[0] = s[1023];
    if (total_out) total_out[0] = (float)s[1023];      // num_edges_kept
  }
}

__global__ __launch_bounds__(256) void k_nscatter(const unsigned* __restrict__ mask,
                                                  const unsigned* __restrict__ boff,
                                                  int* __restrict__ sel, int N) {
  __shared__ unsigned sc[THREADS];
  int t = threadIdx.x, base = blockIdx.x * CHUNK + t * PER_T;
  unsigned c = 0u;
  for (int j = 0; j < PER_T; ++j) { int i = base + j; if (i < N) c += mask[i]; }
  unsigned mine = c;
  sc[t] = c; __syncthreads();
  for (int o = 1; o < THREADS; o <<= 1) {
    unsigned u = (t >= o) ? sc[t - o] : 0u;
    __syncthreads();
    sc[t] += u;
    __syncthreads();
  }
  unsigned pos = boff[blockIdx.x] + sc[t] - mine;
  for (int j = 0; j < PER_T; ++j) {
    int i = base + j;
    if (i < N && mask[i]) sel[pos++] = i;
  }
}

// gather pooled node features: one wave per kept row (128 f32 = 32 lanes x b128)
__global__ __launch_bounds__(256) void k_gather(const float* __restrict__ x,
                                                const int* __restrict__ sel,
                                                const int* __restrict__ region,
                                                const int* __restrict__ batch,
                                                float* __restrict__ xp,
                                                float* __restrict__ rp,
                                                float* __restrict__ bp, int k) {
  int lane = threadIdx.x & 31;
  int w = (int)((blockIdx.x * (blockDim.x >> 5)) + (threadIdx.x >> 5));
  if (w >= k) return;
  int i = sel[w];
  const v4f* src = (const v4f*)(x + (size_t)i * 128);
  v4f* dst = (v4f*)(xp + (size_t)w * 128);
  dst[lane] = src[lane];
  if (lane == 0) { rp[w] = (float)region[i]; bp[w] = (float)batch[i]; }
}

// --------------------- edge stable partition --------------------------------
__global__ __launch_bounds__(256) void k_ecount(const int* __restrict__ ei,
                                                const unsigned* __restrict__ mask,
                                                unsigned* bcnt, int E) {
  __shared__ unsigned red[THREADS];
  const int* e0 = ei; const int* e1 = ei + E;
  int t = threadIdx.x, base = blockIdx.x * CHUNK + t * PER_T;
  unsigned c = 0u;
  for (int j = 0; j < PER_T; ++j) {
    int e = base + j;
    if (e < E) c += mask[e0[e]] & mask[e1[e]];
  }
  red[t] = c; __syncthreads();
  for (int s = THREADS / 2; s > 0; s >>= 1) { if (t < s) red[t] += red[t + s]; __syncthreads(); }
  if (t == 0) bcnt[blockIdx.x] = red[0];
}

// stable partition scatter: kept edge -> front slot (original order),
// dropped edge -> total + (#dropped before it); every slot written each call.
__global__ __launch_bounds__(256) void k_escatter(const int* __restrict__ ei,
                                                  const float* __restrict__ ea,
                                                  const unsigned* __restrict__ mask,
                                                  const unsigned* __restrict__ boff,
                                                  const unsigned* __restrict__ total_ws,
                                                  float* __restrict__ oi,
                                                  float* __restrict__ oa, int E) {
  __shared__ unsigned sc[THREADS];
  const int* e0 = ei; const int* e1 = ei + E;
  int t = threadIdx.x, base = blockIdx.x * CHUNK + t * PER_T;
  unsigned c = 0u;
  for (int j = 0; j < PER_T; ++j) {
    int e = base + j;
    if (e < E) c += mask[e0[e]] & mask[e1[e]];
  }
  unsigned mine = c;
  sc[t] = c; __syncthreads();
  for (int o = 1; o < THREADS; o <<= 1) {
    unsigned u = (t >= o) ? sc[t - o] : 0u;
    __syncthreads();
    sc[t] += u;
    __syncthreads();
  }
  unsigned keptBefore = boff[blockIdx.x] + sc[t] - mine;
  unsigned total = total_ws[0];
  float* oi0 = oi; float* oi1 = oi + E;
  for (int j = 0; j < PER_T; ++j) {
    int e = base + j;
    if (e >= E) break;
    int a = e0[e], b = e1[e];
    unsigned kp = mask[a] & mask[b];
    if (kp) {
      unsigned pos = keptBefore++;
      oi0[pos] = (float)a; oi1[pos] = (float)b; oa[pos] = ea[e];
    } else {
      unsigned pos = total + ((unsigned)e - keptBefore);
      oi0[pos] = -1.0f; oi1[pos] = -1.0f; oa[pos] = 0.0f;
    }
  }
}

// ---------------------------------------------------------------------------
extern "C" void kernel_launch(void* const* d_in, const int* in_sizes, int n_in,
                              void* d_out, int out_size, void* d_ws, size_t ws_size,
                              hipStream_t stream) {
  (void)n_in; (void)out_size; (void)ws_size;
  const float* x      = (const float*)d_in[0];
  const int*   ei     = (const int*)d_in[1];
  const float* ea     = (const float*)d_in[2];
  const int*   region = (const int*)d_in[3];
  const int*   batch  = (const int*)d_in[4];
  const float* W      = (const float*)d_in[5];
  const float* bias   = (const float*)d_in[6];

  int C = in_sizes[5];            // W is [C,1]
  int N = in_sizes[0] / C;
  int E = in_sizes[2];
  int k = N / 2; if (k < 1) k = 1;

  float* out = (float*)d_out;
  size_t off_ei = (size_t)k * C;
  size_t off_ea = off_ei + 2 * (size_t)E;
  size_t off_rp = off_ea + (size_t)E;
  size_t off_bp = off_rp + (size_t)k;
  size_t off_sc = off_bp + (size_t)k;
  float* xp  = out;
  float* oei = out + off_ei;
  float* oea = out + off_ea;
  float* rp  = out + off_rp;
  float* bp  = out + off_bp;
  float* sc  = out + off_sc;
  float* ne  = out + off_sc + (size_t)N;     // num_edges_kept scalar

  unsigned* hist  = (unsigned*)d_ws;         // 2048
  unsigned* ctl   = hist + 2048;             // 16 (prefix,krem,above,tiectr,-,th,ties,-,etotal)
  unsigned* ebcnt = ctl + 16;                // <=1024
  unsigned* eboff = ebcnt + 1024;            // <=1024
  unsigned* nbcnt = eboff + 1024;            // <=1024
  unsigned* nboff = nbcnt + 1024;            // <=1024
  unsigned* maskb = nboff + 1024;            // N
  int*      sel   = (int*)(maskb + N);       // k

  int nbE = (E + CHUNK - 1) / CHUNK;         // 782 for E=3.2M (<=1024 req.)
  int nbN = (N + CHUNK - 1) / CHUNK;

  k_init<<<8, 256, 0, stream>>>(hist, ctl, (unsigned)k);

  int nWaves = (N + 15) / 16;
  k_scores<<<(nWaves + 7) / 8, 256, 0, stream>>>(x, W, bias, sc, N);

  for (int pass = 0; pass < 3; ++pass) {
    k_hist<<<256, 256, 0, stream>>>(sc, hist, ctl, N, pass);
    int nbins = (pass == 2) ? 1024 : 2048;
    int bits  = (pass == 2) ? 10 : 11;
    k_scan_radix<<<1, 1024, 0, stream>>>(hist, ctl, nbins, bits, (pass == 2) ? 1 : 0);
  }

  k_mask<<<(N + 255) / 256, 256, 0, stream>>>(sc, maskb, ctl, N);

  // deterministic stable compaction of kept nodes (ascending node index)
  k_ncount<<<nbN, 256, 0, stream>>>(maskb, nbcnt, N);
  k_scan_blocks<<<1, 1024, 0, stream>>>(nbcnt, nboff, nbN, (unsigned*)nullptr, (float*)nullptr);
  k_nscatter<<<nbN, 256, 0, stream>>>(maskb, nboff, sel, N);
  k_gather<<<(k + 7) / 8, 256, 0, stream>>>(x, sel, region, batch, xp, rp, bp, k);

  // edge stable partition
  k_ecount<<<nbE, 256, 0, stream>>>(ei, maskb, ebcnt, E);
  k_scan_blocks<<<1, 1024, 0, stream>>>(ebcnt, eboff, nbE, ctl + 8, ne);
  k_escatter<<<nbE, 256, 0, stream>>>(ei, ea, maskb, eboff, ctl + 8, oei, oea, E);
}